// ConditionalRandomField_37048387895449
// MI455X (gfx1250) — compile-verified
//
#include <hip/hip_runtime.h>
#include <hip/hip_bf16.h>
#include <cmath>

typedef float v2f __attribute__((ext_vector_type(2)));
typedef float v8f __attribute__((ext_vector_type(8)));

#define CRF_B     512
#define CRF_L     1024
#define CRF_T     64
#define CRF_START 62
#define CRF_STOP  63
#define NGROUPS   (CRF_B / 16)

// ---------------------------------------------------------------------------
// Kernel 1: Epack[idx] = exp(transitions) rearranged into WMMA B-fragment
// lane order.  B-matrix 4x16 f32 layout (mirrors ISA A 16x4 layout):
//   lane = h*16 + ln;  VGPR c (c=0,1) holds K = 4*k + 2*h + c, N = ntile*16+ln
// Epack[((ntile*16 + k)*32 + lane)*2 + c] = exp(T[K][N])
// ---------------------------------------------------------------------------
__global__ void crf_pack_exp(const float* __restrict__ trans,
                             float* __restrict__ epack) {
  int idx = blockIdx.x * blockDim.x + threadIdx.x;   // 0..4095
  if (idx >= CRF_T * CRF_T) return;
  int ntile = idx >> 10;
  int rem   = idx & 1023;
  int k     = rem >> 6;
  int r2    = rem & 63;
  int lane  = r2 >> 1;
  int c     = r2 & 1;
  int h     = lane >> 4;
  int ln    = lane & 15;
  int K = 4 * k + 2 * h + c;
  int N = ntile * 16 + ln;
  epack[idx] = expf(trans[K * CRF_T + N]);
}

// ---------------------------------------------------------------------------
// Kernel 2: forward (log-denominator).  One block = 16 batches, 4 waves,
// wave w owns output N-tile [16w, 16w+16).  Alpha ping-pongs in LDS.
// Per step: P = exp(alpha - rowmax) -> A-frags in registers,
//           S = P @ E via 16x V_WMMA_F32_16X16X4_F32 per wave,
//           alpha' = logits + rowmax + log(S)   (mask-aware, branchless).
// ---------------------------------------------------------------------------
__global__ __launch_bounds__(128) void crf_forward(
    const float* __restrict__ inputs,   // [B][L][T]
    const float* __restrict__ trans,    // [T][T]
    const int*   __restrict__ mask,     // [B][L]
    const float* __restrict__ epack,    // [4096] packed exp(trans)
    float*       __restrict__ denom)    // [B]
{
  __shared__ float alphaBuf[2][16][CRF_T + 2];   // +2 pad: conflict-free banks
  __shared__ float ef[CRF_T * CRF_T];            // packed E fragments
  __shared__ float m_arr[4][16];                 // per-wave rowmax broadcast
  __shared__ float lp_arr[4][16];                // per-wave log(sum p) (mask=0 path)
  __shared__ int   mk_arr[4][16];                // per-wave mask broadcast

  const int g    = blockIdx.x;          // batch group (16 batches)
  const int tid  = threadIdx.x;
  const int wave = tid >> 5;            // = N-tile index
  const int lane = tid & 31;
  const int h    = lane >> 4;           // lane half
  const int ln   = lane & 15;
  const int n0   = wave * 16;

  const float* inB = inputs + (size_t)g * 16 * CRF_L * CRF_T;

  // Stage packed E into LDS (read every step by all waves).
  for (int i = tid; i < CRF_T * CRF_T; i += 128) ef[i] = epack[i];

  // alpha0[b][t] = trans[START][t] + logits[b][0][t]
  for (int i = tid; i < 16 * CRF_T; i += 128) {
    int b = i >> 6, t = i & 63;
    alphaBuf[0][b][t] = trans[CRF_START * CRF_T + t] + inB[(size_t)b * CRF_L * CRF_T + t];
  }
  __syncthreads();

  int cur = 0;
  for (int step = 1; step < CRF_L; ++step) {
    // --- early global loads: logits for this wave's tile, in D-frag layout ---
    float lg[8];
#pragma unroll
    for (int r = 0; r < 8; ++r) {
      int b = r + 8 * h;
      lg[r] = inB[(size_t)b * CRF_L * CRF_T + (size_t)step * CRF_T + n0 + ln];
    }
    if (lane < 16)
      mk_arr[wave][ln] = mask[(size_t)(g * 16 + ln) * CRF_L + (step - 1)];

    // --- read alpha in A-fragment K-pattern; row max over this lane's tags ---
    float av[32];
    float mx = -__builtin_inff();
#pragma unroll
    for (int k = 0; k < 16; ++k) {
      float2 v = *(const float2*)&alphaBuf[cur][ln][4 * k + 2 * h];
      av[2 * k]     = v.x;
      av[2 * k + 1] = v.y;
      mx = fmaxf(mx, fmaxf(v.x, v.y));
    }
    mx = fmaxf(mx, __shfl_xor(mx, 16));          // combine lane halves -> full row max

    // --- P = exp(alpha - mx): lands directly in A-fragments ---
    v2f afr[16];
    float psum = 0.f;
#pragma unroll
    for (int k = 0; k < 16; ++k) {
      float px = __expf(av[2 * k] - mx);
      float py = __expf(av[2 * k + 1] - mx);
      afr[k].x = px;
      afr[k].y = py;
      psum += px + py;
    }
    psum += __shfl_xor(psum, 16);
    if (lane < 16) {
      m_arr[wave][ln]  = mx;
      lp_arr[wave][ln] = __logf(psum);           // used when mask[step-1]==0
    }

    // --- S(tile) = P @ E(tile): 16 chained fp32 WMMA ops ---
    v8f acc = {};
#pragma unroll
    for (int k = 0; k < 16; ++k) {
      float2 bf = *(const float2*)&ef[((wave * 16 + k) * 64 + lane * 2)];
      v2f bfr;
      bfr.x = bf.x;
      bfr.y = bf.y;
      acc = __builtin_amdgcn_wmma_f32_16x16x4_f32(
          /*neg_a=*/false, afr[k], /*neg_b=*/false, bfr,
          /*c_mod=*/(short)0, acc, /*reuse_a=*/false, /*reuse_b=*/false);
    }

    // --- alpha'[b][t] = logit + mx + log S (or mx + log(sum p) if masked) ---
    // Branchless: evaluate both arms, select with v_cndmask (no exec branches).
    int nxt = cur ^ 1;
#pragma unroll
    for (int r = 0; r < 8; ++r) {
      int   b  = r + 8 * h;
      float M  = m_arr[wave][b];
      float vt = lg[r] + M + __logf(acc[r]);     // mask == 1 path
      float vf = M + lp_arr[wave][b];            // mask == 0 path
      float val = (mk_arr[wave][b] != 0) ? vt : vf;
      alphaBuf[nxt][b][n0 + ln] = val;
    }
    __syncthreads();
    cur = nxt;
  }

  // --- denominator: logsumexp(alpha + trans[:,STOP]) per batch ---
  if (wave == 0 && lane < 16) {
    float mx = -__builtin_inff();
    for (int t = 0; t < CRF_T; ++t)
      mx = fmaxf(mx, alphaBuf[cur][ln][t] + trans[t * CRF_T + CRF_STOP]);
    float s = 0.f;
    for (int t = 0; t < CRF_T; ++t)
      s += __expf(alphaBuf[cur][ln][t] + trans[t * CRF_T + CRF_STOP] - mx);
    denom[g * 16 + ln] = mx + __logf(s);
  }
}

// ---------------------------------------------------------------------------
// Kernel 3: joint likelihood (numerator).  One block per batch, deterministic
// LDS tree reduction.
// ---------------------------------------------------------------------------
__global__ __launch_bounds__(64) void crf_numer(
    const float* __restrict__ inputs, const float* __restrict__ trans,
    const long long* __restrict__ tags, const int* __restrict__ mask,
    float* __restrict__ numer)
{
  __shared__ float sred[64];
  __shared__ float mred[64];
  const int b   = blockIdx.x;
  const int tid = threadIdx.x;
  const float*     lb = inputs + (size_t)b * CRF_L * CRF_T;
  const long long* tg = tags + (size_t)b * CRF_L;
  const int*       mk = mask + (size_t)b * CRF_L;

  float s = 0.f, ms = 0.f;
  for (int i = tid; i < CRF_L; i += 64) {
    float mfi = (float)mk[i];
    ms += mfi;
    if (i < CRF_L - 1) {
      int ti = (int)tg[i];
      int tj = (int)tg[i + 1];
      float mfn = (float)mk[i + 1];
      s += trans[ti * CRF_T + tj] * mfn + lb[(size_t)i * CRF_T + ti] * mfi;
    }
  }
  sred[tid] = s;
  mred[tid] = ms;
  __syncthreads();
  for (int st = 32; st > 0; st >>= 1) {
    if (tid < st) {
      sred[tid] += sred[tid + st];
      mred[tid] += mred[tid + st];
    }
    __syncthreads();
  }
  if (tid == 0) {
    int   last = (int)mred[0] - 1;
    int   t0   = (int)tg[0];
    int   lt   = (int)tg[last];
    float mfL  = (float)mk[CRF_L - 1];
    numer[b] = sred[0] + trans[CRF_START * CRF_T + t0] + trans[lt * CRF_T + CRF_STOP]
             + lb[(size_t)(CRF_L - 1) * CRF_T + lt] * mfL;
  }
}

// ---------------------------------------------------------------------------
// Kernel 4: fixed-order scalar reduction (deterministic, no float atomics).
// ---------------------------------------------------------------------------
__global__ void crf_reduce(const float* __restrict__ numer,
                           const float* __restrict__ denom,
                           float* __restrict__ out) {
  if (blockIdx.x == 0 && threadIdx.x == 0) {
    float t = 0.f;
    for (int b = 0; b < CRF_B; ++b) t += numer[b] - denom[b];
    out[0] = t;
  }
}

extern "C" void kernel_launch(void* const* d_in, const int* in_sizes, int n_in,
                              void* d_out, int out_size, void* d_ws, size_t ws_size,
                              hipStream_t stream) {
  const float*     inputs = (const float*)d_in[0];       // [512][1024][64] f32
  const float*     trans  = (const float*)d_in[1];       // [64][64] f32
  const long long* tags   = (const long long*)d_in[2];   // [512][1024] i64
  const int*       mask   = (const int*)d_in[3];         // [512][1024] i32

  float* ws    = (float*)d_ws;
  float* epack = ws;                 // 4096 floats
  float* denom = ws + 4096;          // 512 floats
  float* numer = ws + 4096 + 512;    // 512 floats

  crf_pack_exp<<<16, 256, 0, stream>>>(trans, epack);
  crf_forward<<<NGROUPS, 128, 0, stream>>>(inputs, trans, mask, epack, denom);
  crf_numer<<<CRF_B, 64, 0, stream>>>(inputs, trans, tags, mask, numer);
  crf_reduce<<<1, 32, 0, stream>>>(numer, denom, (float*)d_out);
}